// CTCLossLayer_40518721470858
// MI455X (gfx1250) — compile-verified
//
#include <hip/hip_runtime.h>
#include <math.h>

typedef __attribute__((ext_vector_type(2))) float v2f;
typedef __attribute__((ext_vector_type(8))) float v8f;

#define B_  64
#define T_  2000
#define V_  128
#define L_  200
#define S_  401           // 2*L + 1
#define NEGF (-1e30f)
#define PEN  25.0f

#if defined(__has_builtin)
#if __has_builtin(__builtin_amdgcn_wmma_f32_16x16x4_f32)
#define HAVE_WMMA_F32X4 1
#endif
#endif

// ---------------------------------------------------------------------------
// Kernel 1: norm[row] = log( sum_v exp(y[row,v] - PEN*[v==0]) ), row = b*T + t
// One wave handles 16 rows. exp row-sums are reduced with chained
// V_WMMA_F32_16X16X4_F32 against an all-ones B matrix (D = sum_k A + C).
// A layout (32-bit, 16x4): lanes 0-15 supply K0,K1 of row=lane,
//                          lanes 16-31 supply K2,K3 of row=lane-16.
// ---------------------------------------------------------------------------
__global__ __launch_bounds__(256) void ctc_norm_kernel(const float* __restrict__ y,
                                                       float* __restrict__ norm) {
  const int lane = threadIdx.x & 31;
  const int wave = threadIdx.x >> 5;
  const int row_base = (blockIdx.x * 8 + wave) * 16;   // 16 rows per wave
  const int r  = lane & 15;                             // row within tile
  const int kp = lane >> 4;                             // 0: K0,K1   1: K2,K3
  const float* rowp = y + (size_t)(row_base + r) * V_ + 2 * kp;

#ifdef HAVE_WMMA_F32X4
  v8f acc = {0.f, 0.f, 0.f, 0.f, 0.f, 0.f, 0.f, 0.f};
  const v2f ones = {1.0f, 1.0f};
#pragma unroll
  for (int c = 0; c < 32; ++c) {
    v2f x = *(const v2f*)(rowp + 4 * c);
    // blank penalty hits column v==0 only: chunk 0, lane-half 0, first elem
    x.x -= ((c == 0) && (kp == 0)) ? PEN : 0.0f;
    v2f a;
    a.x = __expf(x.x);
    a.y = __expf(x.y);
    // D = A(16x4) x ones(4x16) + C   => every column of D = rowsum(A)
    acc = __builtin_amdgcn_wmma_f32_16x16x4_f32(false, a, false, ones,
                                                (short)0, acc, false, false);
  }
  // C/D layout: VGPR i, lanes 0-15 -> M=i ; lanes 16-31 -> M=i+8
  if (lane == 0) {
#pragma unroll
    for (int i = 0; i < 8; ++i) norm[row_base + i] = __logf(acc[i]);
  } else if (lane == 16) {
#pragma unroll
    for (int i = 0; i < 8; ++i) norm[row_base + 8 + i] = __logf(acc[i]);
  }
#else
  // Fallback: per-lane partial sums + cross-half shuffle combine
  float p = 0.0f;
#pragma unroll
  for (int c = 0; c < 32; ++c) {
    v2f x = *(const v2f*)(rowp + 4 * c);
    x.x -= ((c == 0) && (kp == 0)) ? PEN : 0.0f;
    p += __expf(x.x) + __expf(x.y);
  }
  float other = __shfl_xor(p, 16, 32);
  if (lane < 16) norm[row_base + r] = __logf(p + other);
#endif
}

// ---------------------------------------------------------------------------
// Kernel 2: alpha recursion. One workgroup (8 wave32) per batch element.
// alpha kept in LDS (double buffer); one 128-float logit row staged in LDS
// per step; next step's row prefetched via global_prefetch.
// ---------------------------------------------------------------------------
__global__ __launch_bounds__(256) void ctc_alpha_kernel(const float* __restrict__ y,
                                                        const int* __restrict__ labels,
                                                        const float* __restrict__ norm,
                                                        float* __restrict__ losses) {
  const int b   = blockIdx.x;
  const int tid = threadIdx.x;

  __shared__ int   ext[S_ + 3];
  __shared__ int   allow[S_ + 3];
  __shared__ float aA[S_ + 3];
  __shared__ float aB[S_ + 3];
  __shared__ float rowbuf[V_];
  __shared__ float normv;
  __shared__ int   cnt;

  // extended label sequence: blank, l1, blank, l2, ..., blank
  for (int s = tid; s < S_; s += 256)
    ext[s] = (s & 1) ? labels[b * L_ + (s >> 1)] : 0;
  if (tid == 0) cnt = 0;
  __syncthreads();
  for (int s = tid; s < S_; s += 256)
    allow[s] = (s >= 2) && (ext[s] != 0) && (ext[s] != ext[s - 2]);
  if (tid < L_) { if (labels[b * L_ + tid] != 0) atomicAdd(&cnt, 1); }

  // stage t = 0 row
  if (tid < V_) rowbuf[tid] = y[(size_t)b * T_ * V_ + tid];
  if (tid == 0) normv = norm[b * T_];
  __syncthreads();

  float* ap = aA;
  float* ac = aB;
  // alpha0
  for (int s = tid; s < S_; s += 256) {
    int   e  = ext[s];
    float lp = rowbuf[e] - ((e == 0) ? PEN : 0.0f) - normv;
    ap[s] = (s <= 1) ? lp : NEGF;
  }

  for (int t = 1; t < T_; ++t) {
    __syncthreads();   // alpha(t-1) writes done; rowbuf(t-1) reads done
    if (tid < V_) {
      const size_t off = ((size_t)b * T_ + t) * V_ + tid;
      rowbuf[tid] = y[off];
      if (t + 1 < T_) __builtin_prefetch(y + off + V_, 0, 1);  // global_prefetch_b8
    }
    if (tid == 0) normv = norm[b * T_ + t];
    __syncthreads();

    const float nv = normv;
#pragma unroll
    for (int pass = 0; pass < 2; ++pass) {
      const int s = tid + pass * 256;
      if (s < S_) {
        float a0 = ap[s];
        float a1 = (s >= 1)  ? ap[(s >= 1) ? s - 1 : 0] : NEGF;
        float a2 = allow[s]  ? ap[(s >= 2) ? s - 2 : 0] : NEGF;
        float m  = fmaxf(a0, fmaxf(a1, a2));
        float sm = __expf(a0 - m) + __expf(a1 - m) + __expf(a2 - m);
        int   e  = ext[s];
        float lp = rowbuf[e] - ((e == 0) ? PEN : 0.0f) - nv;
        ac[s] = m + __logf(sm) + lp;
      }
    }
    float* tmp = ap; ap = ac; ac = tmp;
  }
  __syncthreads();

  if (tid == 0) {
    int   ll = cnt;
    int   i1 = 2 * ll;
    int   i2 = (2 * ll - 1 > 0) ? (2 * ll - 1) : 0;
    float x1 = ap[i1];
    float x2 = ap[i2];
    float m  = fmaxf(x1, x2);
    losses[b] = -(m + __logf(__expf(x1 - m) + __expf(x2 - m)));
  }
}

// ---------------------------------------------------------------------------
// Kernel 3: out = mean(losses) + 1e-7
// ---------------------------------------------------------------------------
__global__ __launch_bounds__(64) void ctc_reduce_kernel(const float* __restrict__ losses,
                                                        float* __restrict__ out) {
  __shared__ float sdata[64];
  const int tid = threadIdx.x;
  sdata[tid] = losses[tid];
  __syncthreads();
  for (int off = 32; off > 0; off >>= 1) {
    if (tid < off) sdata[tid] += sdata[tid + off];
    __syncthreads();
  }
  if (tid == 0) out[0] = sdata[0] * (1.0f / B_) + 1e-7f;
}

extern "C" void kernel_launch(void* const* d_in, const int* in_sizes, int n_in,
                              void* d_out, int out_size, void* d_ws, size_t ws_size,
                              hipStream_t stream) {
  const float* y      = (const float*)d_in[0];   // [B,T,V] fp32
  const int*   labels = (const int*)d_in[1];     // [B,L] int32
  float* norm   = (float*)d_ws;                  // B*T floats (512 KB)
  float* losses = norm + (size_t)B_ * T_;        // 64 floats

  // 128000 rows / (8 waves * 16 rows) = 1000 blocks
  ctc_norm_kernel<<<(B_ * T_) / 128, 256, 0, stream>>>(y, norm);
  ctc_alpha_kernel<<<B_, 256, 0, stream>>>(y, labels, norm, losses);
  ctc_reduce_kernel<<<1, 64, 0, stream>>>(losses, (float*)d_out);
}